// Sparse3DCNN_15650860827382
// MI455X (gfx1250) — compile-verified
//
#include <hip/hip_runtime.h>

#define K_ADA 0.1f
#define EPS_AN 1e-5f

typedef __attribute__((ext_vector_type(2))) float v2f;
typedef __attribute__((ext_vector_type(8))) float v8f;

// ---------------- mask downsample: 3x3x3 max, stride 2, pad 1 ----------------
__global__ void mask_down_kernel(const float* __restrict__ min_, float* __restrict__ mout,
                                 int DIN, int nout) {
  const int DOUT = DIN >> 1;
  int i = blockIdx.x * blockDim.x + threadIdx.x;
  if (i >= nout) return;
  const int d3 = DOUT * DOUT * DOUT;
  int b = i / d3;
  int r = i - b * d3;
  int z = r / (DOUT * DOUT); r -= z * (DOUT * DOUT);
  int y = r / DOUT;
  int x = r - y * DOUT;
  float mx = 0.f;
  for (int dz = -1; dz <= 1; ++dz) {
    int iz = 2 * z + dz; if ((unsigned)iz >= (unsigned)DIN) continue;
    for (int dy = -1; dy <= 1; ++dy) {
      int iy = 2 * y + dy; if ((unsigned)iy >= (unsigned)DIN) continue;
      for (int dx = -1; dx <= 1; ++dx) {
        int ix = 2 * x + dx; if ((unsigned)ix >= (unsigned)DIN) continue;
        mx = fmaxf(mx, min_[((b * DIN + iz) * DIN + iy) * DIN + ix]);
      }
    }
  }
  mout[i] = mx;
}

// ------ layer 1 (CIN=1): direct per-voxel conv + fused AdaNorm/ReLU/mask ------
// One thread = one voxel; all 16 channels thread-private -> shuffle-free norm,
// contiguous 64B stores. Taps are compile-time constants (no div/mod chains).
__global__ __launch_bounds__(256) void conv1_an_kernel(
    const float* __restrict__ xin, const float* __restrict__ wgt,   // [27][16]
    const float* __restrict__ bias, const float* __restrict__ gamma,
    const float* __restrict__ mask, float* __restrict__ out, int nvox) {
  const int i = blockIdx.x * blockDim.x + threadIdx.x;
  if (i >= nvox) return;
  float* orow = out + (long)i * 16;
  const float mv = mask[i];
  if (mv == 0.f) {  // inactive voxel: next layer's im2col needs zeros
#pragma unroll
    for (int co = 0; co < 16; ++co) orow[co] = 0.f;
    return;
  }
  constexpr int DIN = 96;
  constexpr int D3 = DIN * DIN * DIN;
  const int b = i / D3;
  int r = i - b * D3;
  const int z = r / (DIN * DIN); r -= z * (DIN * DIN);
  const int y = r / DIN;
  const int x = r - y * DIN;

  float v[27];
#pragma unroll
  for (int tap = 0; tap < 27; ++tap) {
    const int kz = tap / 9, ky = (tap / 3) % 3, kx = tap % 3;  // constants
    const int iz = z + kz - 1, iy = y + ky - 1, ix = x + kx - 1;
    const bool inb = (unsigned)iz < (unsigned)DIN && (unsigned)iy < (unsigned)DIN &&
                     (unsigned)ix < (unsigned)DIN;
    const int izc = inb ? iz : 0, iyc = inb ? iy : 0, ixc = inb ? ix : 0;
    const float t = xin[((b * DIN + izc) * DIN + iyc) * DIN + ixc];
    v[tap] = inb ? t : 0.f;
  }
  float o[16];
  float s = 0.f;
#pragma unroll
  for (int co = 0; co < 16; ++co) {
    float acc = bias[co];
#pragma unroll
    for (int tap = 0; tap < 27; ++tap) acc = fmaf(v[tap], wgt[tap * 16 + co], acc);
    o[co] = acc * mv;
    s += o[co];
  }
  const float mean = s * (1.f / 16.f);
  float q = 0.f;
#pragma unroll
  for (int co = 0; co < 16; ++co) {
    const float d = o[co] - mean;
    q += d * d;
  }
  const float rstd = rsqrtf(q * (1.f / 16.f) + EPS_AN);
#pragma unroll
  for (int co = 0; co < 16; ++co) {
    const float xn = (o[co] - mean) * rstd;
    orow[co] = fmaxf(gamma[co] * (1.f - K_ADA * xn) * xn, 0.f) * mv;
  }
}

// ------ layers 2-5: conv3d 3x3x3 (pad 1) as implicit GEMM via f32 WMMA, ------
// ------ fused AdaNorm+ReLU+mask epilogue (full Cout wave-resident)       ------
template <int CIN, int COUT, int STRIDE, int DIN>
__global__ __launch_bounds__(128) void conv_an_wmma_kernel(
    const float* __restrict__ xin, const float* __restrict__ wgt,
    const float* __restrict__ bias, const float* __restrict__ gamma,
    const float* __restrict__ mout, float* __restrict__ out) {
  static_assert((CIN & 3) == 0, "tap-major path requires CIN % 4 == 0");
  constexpr int DOUT = DIN / STRIDE;
  constexpr int D3 = DOUT * DOUT * DOUT;
  constexpr int NT = COUT / 16;

  const int lane = threadIdx.x & 31;
  const int wv   = threadIdx.x >> 5;   // 4 waves per block
  const int half = lane >> 4;          // selects K pair {0,1} vs {2,3}
  const int l    = lane & 15;          // A row / B column within tile
  const int m0   = (blockIdx.x * 4 + wv) * 16;

  // Tile-level sparsity skip (wave-uniform)
  const bool rowact = (mout[m0 + l] > 0.f);
  if (__ballot(rowact) == 0ull) {
#pragma unroll
    for (int rr = 0; rr < 8; ++rr)
#pragma unroll
      for (int nt = 0; nt < NT; ++nt)
        out[(long)(m0 + rr + 8 * half) * COUT + nt * 16 + l] = 0.f;
    return;
  }

  // Decompose this lane's A-row voxel -> (b, zo, yo, xo)
  const int vm = m0 + l;
  const int b  = vm / D3;
  int rem = vm - b * D3;
  const int zo = rem / (DOUT * DOUT); rem -= zo * (DOUT * DOUT);
  const int yo = rem / DOUT;
  const int xo = rem - yo * DOUT;

  v8f c[NT];
#pragma unroll
  for (int nt = 0; nt < NT; ++nt) c[nt] = (v8f){};

#pragma unroll 1
  for (int tap = 0; tap < 27; ++tap) {
    const int kz = tap / 9;
    const int ky = (tap / 3) % 3;
    const int kx = tap % 3;
    const int iz = zo * STRIDE + kz - 1;
    const int iy = yo * STRIDE + ky - 1;
    const int ix = xo * STRIDE + kx - 1;
    const bool inb = (unsigned)iz < (unsigned)DIN && (unsigned)iy < (unsigned)DIN &&
                     (unsigned)ix < (unsigned)DIN;
    // clamped always-valid address + post-load select: keeps EXEC uniform,
    // no exec-save/branch pairs in the inner loop
    const int izc = iz < 0 ? 0 : (iz >= DIN ? DIN - 1 : iz);
    const int iyc = iy < 0 ? 0 : (iy >= DIN ? DIN - 1 : iy);
    const int ixc = ix < 0 ? 0 : (ix >= DIN ? DIN - 1 : ix);
    const float* arow = xin + (long)(((b * DIN + izc) * DIN + iyc) * DIN + ixc) * CIN;
    const float* brow = wgt + (long)(tap * CIN) * COUT + l;
    if (tap + 1 < 27)  // gfx1250 global_prefetch_b8 of next tap's weights
      __builtin_prefetch(wgt + (long)((tap + 1) * CIN) * COUT + l, 0, 3);
#pragma unroll
    for (int ci = 0; ci < CIN; ci += 4) {
      v2f a = *(const v2f*)(arow + ci + 2 * half);  // 8B-aligned pair load
      a[0] = inb ? a[0] : 0.f;
      a[1] = inb ? a[1] : 0.f;
#pragma unroll
      for (int nt = 0; nt < NT; ++nt) {
        v2f bb;
        bb[0] = brow[(ci + 2 * half + 0) * COUT + nt * 16];
        bb[1] = brow[(ci + 2 * half + 1) * COUT + nt * 16];
        c[nt] = __builtin_amdgcn_wmma_f32_16x16x4_f32(
            false, a, false, bb, (short)0, c[nt], false, false);
      }
    }
  }

  // ---- fused epilogue: bias -> mask -> AdaNorm -> ReLU -> mask ----
  float bsl[NT], gml[NT];
#pragma unroll
  for (int nt = 0; nt < NT; ++nt) {
    bsl[nt] = bias[nt * 16 + l];
    gml[nt] = gamma[nt * 16 + l];
  }
#pragma unroll
  for (int rr = 0; rr < 8; ++rr) {
    const int mr = m0 + rr + 8 * half;
    const float mv = mout[mr];
    float v[NT];
    float s = 0.f;
#pragma unroll
    for (int nt = 0; nt < NT; ++nt) {
      v[nt] = (c[nt][rr] + bsl[nt]) * mv;
      s += v[nt];
    }
    // lanes 0..15 hold voxel m0+rr, lanes 16..31 hold m0+rr+8;
    // xor offsets {8,4,2,1} stay within each 16-lane half on wave32
#pragma unroll
    for (int off = 8; off >= 1; off >>= 1) s += __shfl_xor(s, off);
    const float mean = s * (1.f / COUT);
    float q = 0.f;
#pragma unroll
    for (int nt = 0; nt < NT; ++nt) {
      const float d = v[nt] - mean;
      q += d * d;
    }
#pragma unroll
    for (int off = 8; off >= 1; off >>= 1) q += __shfl_xor(q, off);
    const float rstd = rsqrtf(q * (1.f / COUT) + EPS_AN);
#pragma unroll
    for (int nt = 0; nt < NT; ++nt) {
      const float xn = (v[nt] - mean) * rstd;
      const float y = gml[nt] * (1.f - K_ADA * xn) * xn;
      out[(long)mr * COUT + nt * 16 + l] = fmaxf(y, 0.f) * mv;
    }
  }
}

// ---------------- global max pool over active voxels; one block per (b, c) ----------------
__global__ __launch_bounds__(256) void gmaxpool_kernel(
    const float* __restrict__ h, const float* __restrict__ m3,
    float* __restrict__ pooled, int nvox_per_b, int C) {
  const int bidx = blockIdx.x / C;
  const int cch  = blockIdx.x % C;
  const float* hb = h + (long)bidx * nvox_per_b * C;
  const float* mb = m3 + (long)bidx * nvox_per_b;
  float mx = -3.0e38f;
  for (int v = threadIdx.x; v < nvox_per_b; v += blockDim.x)
    mx = fmaxf(mx, (mb[v] > 0.f) ? hb[(long)v * C + cch] : -3.0e38f);
  __shared__ float sm[256];
  sm[threadIdx.x] = mx;
  __syncthreads();
  for (int s = 128; s > 0; s >>= 1) {
    if (threadIdx.x < s) sm[threadIdx.x] = fmaxf(sm[threadIdx.x], sm[threadIdx.x + s]);
    __syncthreads();
  }
  if (threadIdx.x == 0) pooled[blockIdx.x] = sm[0];
}

// ---------------- head: Linear(64->64) + AdaNorm + ReLU; single tiny block ----------------
__global__ __launch_bounds__(128) void head_kernel(
    const float* __restrict__ pooled, const float* __restrict__ wh,
    const float* __restrict__ bh, const float* __restrict__ gh,
    float* __restrict__ outp) {
  __shared__ float val[128];  // [2][64]
  const int t  = threadIdx.x;
  const int rw = t >> 6;
  const int cc = t & 63;
  float acc = bh[cc];
  for (int k = 0; k < 64; ++k) acc += pooled[rw * 64 + k] * wh[k * 64 + cc];
  val[t] = acc;
  __syncthreads();
  float mean = 0.f;
  for (int k = 0; k < 64; ++k) mean += val[rw * 64 + k];
  mean *= (1.f / 64.f);
  float var = 0.f;
  for (int k = 0; k < 64; ++k) {
    const float d = val[rw * 64 + k] - mean;
    var += d * d;
  }
  var = var * (1.f / 64.f) + EPS_AN;
  const float xn = (acc - mean) * rsqrtf(var);
  const float y = gh[cc] * (1.f - K_ADA * xn) * xn;
  outp[t] = fmaxf(y, 0.f);
}

// ---------------- launcher ----------------
extern "C" void kernel_launch(void* const* d_in, const int* in_sizes, int n_in,
                              void* d_out, int out_size, void* d_ws, size_t ws_size,
                              hipStream_t stream) {
  (void)in_sizes; (void)n_in; (void)out_size; (void)ws_size;
  const float* x    = (const float*)d_in[0];
  const float* m1   = (const float*)d_in[1];
  const float* w1   = (const float*)d_in[2];
  const float* b1   = (const float*)d_in[3];
  const float* g1   = (const float*)d_in[4];
  const float* w2   = (const float*)d_in[5];
  const float* b2   = (const float*)d_in[6];
  const float* g2   = (const float*)d_in[7];
  const float* w3   = (const float*)d_in[8];
  const float* b3   = (const float*)d_in[9];
  const float* g3   = (const float*)d_in[10];
  const float* w4   = (const float*)d_in[11];
  const float* b4   = (const float*)d_in[12];
  const float* g4   = (const float*)d_in[13];
  const float* w5   = (const float*)d_in[14];
  const float* b5   = (const float*)d_in[15];
  const float* g5   = (const float*)d_in[16];
  const float* wh   = (const float*)d_in[17];
  const float* bh   = (const float*)d_in[18];
  const float* gh   = (const float*)d_in[19];
  float* outp = (float*)d_out;

  const int N1 = 2 * 96 * 96 * 96;
  const int N2 = 2 * 48 * 48 * 48;
  const int N3 = 2 * 24 * 24 * 24;

  // Workspace layout (floats); bufA / bufB ping-pong the activations.
  float* ws     = (float*)d_ws;
  float* m2     = ws;                      // N2
  float* m3     = m2 + N2;                 // N3
  float* pooled = m3 + N3;                 // 128
  float* bufA   = pooled + 128;            // up to N1*16 floats (h1, h3, h5)
  float* bufB   = bufA + (size_t)N1 * 16;  // up to N2*32 floats (h2, h4)

  mask_down_kernel<<<dim3((N2 + 255) / 256), dim3(256), 0, stream>>>(m1, m2, 96, N2);
  mask_down_kernel<<<dim3((N3 + 255) / 256), dim3(256), 0, stream>>>(m2, m3, 48, N3);

  // L1: SubMConv 1->16 @96 (direct VALU kernel, fused AdaNorm/ReLU/mask)
  conv1_an_kernel<<<dim3((N1 + 255) / 256), dim3(256), 0, stream>>>(
      x, w1, b1, g1, m1, bufA, N1);
  // L2: SparseConv s=2, 16->32, 96->48 (WMMA)
  conv_an_wmma_kernel<16, 32, 2, 96>
      <<<dim3(N2 / 64), dim3(128), 0, stream>>>(bufA, w2, b2, g2, m2, bufB);
  // L3: SubMConv 32->32 @48 (WMMA)
  conv_an_wmma_kernel<32, 32, 1, 48>
      <<<dim3(N2 / 64), dim3(128), 0, stream>>>(bufB, w3, b3, g3, m2, bufA);
  // L4: SparseConv s=2, 32->64, 48->24 (WMMA)
  conv_an_wmma_kernel<32, 64, 2, 48>
      <<<dim3(N3 / 64), dim3(128), 0, stream>>>(bufA, w4, b4, g4, m3, bufB);
  // L5: SubMConv 64->64 @24 (WMMA)
  conv_an_wmma_kernel<64, 64, 1, 24>
      <<<dim3(N3 / 64), dim3(128), 0, stream>>>(bufB, w5, b5, g5, m3, bufA);

  // Global max pool over active voxels -> pooled[2][64]
  gmaxpool_kernel<<<dim3(2 * 64), dim3(256), 0, stream>>>(bufA, m3, pooled, 24 * 24 * 24, 64);

  // Head
  head_kernel<<<dim3(1), dim3(128), 0, stream>>>(pooled, wh, bh, gh, outp);
}